// RNNLayerRTRL_13262859010352
// MI455X (gfx1250) — compile-verified
//
#include <hip/hip_runtime.h>
#include <math.h>

#define H 512

typedef float v2f __attribute__((ext_vector_type(2)));
typedef float v4f __attribute__((ext_vector_type(4)));
typedef float v8f __attribute__((ext_vector_type(8)));

// ---------------------------------------------------------------------------
// Kernel 0: th[i] = tanh(h_prev[i])
// ---------------------------------------------------------------------------
__global__ void k_tanh(const float* __restrict__ in, float* __restrict__ out, int n) {
  int i = blockIdx.x * blockDim.x + threadIdx.x;
  if (i < n) out[i] = tanhf(in[i]);
}

// ---------------------------------------------------------------------------
// Kernel 1/2: out = W0 @ v0 + W1 @ v1 (+ bias + pert), optional tanh(out) tap.
// Uses V_WMMA_F32_16X16X4_F32 (f32-exact). One wave computes 16 output rows.
//
// Trick: B[k,n] = v[k] for all n (vector broadcast across columns), so
// D[m,n] = sum_k A[m,k]*v[k] for every column n — each half-wave ends with the
// 8 row results replicated across all its lanes.
//
// f32 A 16x4 layout (ISA 7.12.2): lanes 0-15 hold M=lane, VGPR0=K0, VGPR1=K1;
// lanes 16-31 hold M=lane-16, VGPR0=K2, VGPR1=K3.  B mirrors: low-lane half
// carries the low-K rows.  So per lane: koff = (lane>=16) ? 2 : 0.
// ---------------------------------------------------------------------------
__global__ void k_matvec2_wmma(const float* __restrict__ W0, const float* __restrict__ v0,
                               const float* __restrict__ W1, const float* __restrict__ v1,
                               const float* __restrict__ bias, const float* __restrict__ pert,
                               float* __restrict__ out, float* __restrict__ tanh_out) {
  const int lane     = threadIdx.x & 31;
  const int wave     = blockIdx.x * (blockDim.x >> 5) + (threadIdx.x >> 5);
  const int row_base = wave << 4;                 // 16 rows per wave
  const int row      = row_base + (lane & 15);
  const int koff     = (lane & 16) ? 2 : 0;

  v8f c = {0.f, 0.f, 0.f, 0.f, 0.f, 0.f, 0.f, 0.f};

  const float* a0 = W0 + (size_t)row * H + koff;
  for (int kb = 0; kb < H; kb += 4) {
    v2f a = *(const v2f*)(a0 + kb);
    v2f b = *(const v2f*)(v0 + kb + koff);
    c = __builtin_amdgcn_wmma_f32_16x16x4_f32(false, a, false, b, (short)0, c, false, false);
  }
  const float* a1 = W1 + (size_t)row * H + koff;
  for (int kb = 0; kb < H; kb += 4) {
    v2f a = *(const v2f*)(a1 + kb);
    v2f b = *(const v2f*)(v1 + kb + koff);
    c = __builtin_amdgcn_wmma_f32_16x16x4_f32(false, a, false, b, (short)0, c, false, false);
  }

  // lanes 0-15 hold rows row_base+0..7 in c[0..7]; lanes 16-31 hold +8..15.
  if ((lane & 15) == 0) {
    const int rb = row_base + ((lane & 16) ? 8 : 0);
    for (int p = 0; p < 8; ++p) {
      float r = c[p];
      if (bias) r += bias[rb + p];
      if (pert) r += pert[rb + p];
      out[rb + p] = r;
      if (tanh_out) tanh_out[rb + p] = tanhf(r);
    }
  }
}

// ---------------------------------------------------------------------------
// Kernel 3: jac_b[i,j] = (i==j); dynamics[i,j] = W_hh[i,j] * (1 - th[j]^2).
// One float4 (j-quad) per thread; 512*128 = 65536 threads exactly.
// ---------------------------------------------------------------------------
__global__ void k_small_mats(const float* __restrict__ W_hh, const float* __restrict__ th,
                             float* __restrict__ jac_b, float* __restrict__ dyn) {
  const int t  = blockIdx.x * blockDim.x + threadIdx.x;  // 0..65535
  const int b4 = t & 127;        // j / 4
  const int i  = t >> 7;         // row

  v4f w  = ((const v4f*)W_hh)[t];
  v4f tv = ((const v4f*)th)[b4];
  v4f d  = w * (1.0f - tv * tv);

  v4f e = {0.f, 0.f, 0.f, 0.f};
  const int jb = b4 << 2;
  if (i >= jb && i < jb + 4) e[i - jb] = 1.0f;

  __builtin_nontemporal_store(e, (v4f*)jac_b + t);
  __builtin_nontemporal_store(d, (v4f*)dyn + t);
}

// ---------------------------------------------------------------------------
// Kernel 4: the 512 MB Jacobian fill.  dst[i*H*H + a*H + b] = (i==a)?vec[b]:0.
// Grid-stride over 33,554,432 float4s, 128-bit non-temporal stores (don't let
// 1 GB of write-once data thrash the 192 MB L2).  This is the entire roofline:
// 2 x 512 MB stores at 23.3 TB/s ~ 44 us.
// ---------------------------------------------------------------------------
__global__ void k_jac_fill(const float* __restrict__ vec, float* __restrict__ dst) {
  const v4f* __restrict__ v4 = (const v4f*)vec;
  v4f* __restrict__ d4       = (v4f*)dst;
  const long long n      = (long long)H * H * (H / 4);  // 33,554,432
  const long long stride = (long long)gridDim.x * blockDim.x;
  for (long long t = (long long)blockIdx.x * blockDim.x + threadIdx.x; t < n; t += stride) {
    const int b4 = (int)(t & (H / 4 - 1));   // b quad
    const long long r = t >> 7;              // i*H + a
    const int a = (int)(r & (H - 1));
    const int i = (int)(r >> 9);
    v4f val = {0.f, 0.f, 0.f, 0.f};
    if (i == a) val = v4[b4];
    __builtin_nontemporal_store(val, d4 + t);
  }
}

// ---------------------------------------------------------------------------
extern "C" void kernel_launch(void* const* d_in, const int* in_sizes, int n_in,
                              void* d_out, int out_size, void* d_ws, size_t ws_size,
                              hipStream_t stream) {
  (void)in_sizes; (void)n_in; (void)out_size; (void)ws_size;

  const float* h_prev = (const float*)d_in[0];
  const float* x      = (const float*)d_in[1];
  const float* pert   = (const float*)d_in[2];
  const float* W_hh   = (const float*)d_in[3];
  const float* W_ih   = (const float*)d_in[4];
  const float* b      = (const float*)d_in[5];
  const float* C_w    = (const float*)d_in[6];
  const float* D_w    = (const float*)d_in[7];

  float* out = (float*)d_out;
  float* th  = (float*)d_ws;   // tanh(h_prev), 512 floats
  float* th2 = th + H;         // tanh(h_out),  512 floats

  // Output tuple flat layout (return order):
  float* h_out = out;                               // [H]
  float* y_out = out + H;                           // [H]
  float* jWhh  = out + 2 * H;                       // [H,H,H]
  float* jWih  = jWhh + (long long)H * H * H;       // [H,H,H]
  float* jb    = jWih + (long long)H * H * H;       // [H,H]
  float* dyn   = jb + (long long)H * H;             // [H,H]

  // 0) th = tanh(h_prev)
  k_tanh<<<2, 256, 0, stream>>>(h_prev, th, H);

  // 1) h_out = W_hh@th + W_ih@x + b + pert; th2 = tanh(h_out)
  //    32 waves total (16 rows each) -> 8 blocks x 128 threads (4 waves)
  k_matvec2_wmma<<<8, 128, 0, stream>>>(W_hh, th, W_ih, x, b, pert, h_out, th2);

  // 2) y_out = C_w@th2 + D_w@x
  k_matvec2_wmma<<<8, 128, 0, stream>>>(C_w, th2, D_w, x, nullptr, nullptr, y_out, nullptr);

  // 3) jac_b = I; dynamics = W_hh * (1 - th^2)
  k_small_mats<<<256, 256, 0, stream>>>(W_hh, th, jb, dyn);

  // 4) the two 512 MB Jacobians (pure streaming stores)
  k_jac_fill<<<4096, 256, 0, stream>>>(th, jWhh);
  k_jac_fill<<<4096, 256, 0, stream>>>(x,  jWih);
}